// SAGEEncoder_64854006170164
// MI455X (gfx1250) — compile-verified
//
#include <hip/hip_runtime.h>

#define N_NODES 100000
#define N_EDGES 1600000
#define F 64

typedef __attribute__((ext_vector_type(2))) float v2f;
typedef __attribute__((ext_vector_type(8))) float v8f;

// One thread per (edge, feature). Scatter-add gathered source row into the
// destination accumulator; feature-0 lane also bumps the degree counter
// (layer 1 only -- degree is identical for both layers).
__global__ void sage_scatter_kernel(const float* __restrict__ h,
                                    const int* __restrict__ src,
                                    const int* __restrict__ dst,
                                    float* __restrict__ summed,
                                    float* __restrict__ cnt,
                                    int do_cnt) {
    long long idx = (long long)blockIdx.x * blockDim.x + threadIdx.x;
    int e = (int)(idx >> 6);
    int f = (int)(idx & 63);
    if (e >= N_EDGES) return;
    int s = src[e];
    int d = dst[e];
    atomicAdd(&summed[(long long)d * F + f], h[(long long)s * F + f]);
    if (do_cnt && f == 0) atomicAdd(&cnt[d], 1.0f);
}

// One wave computes a 16-node x 16-output tile of
//   out = (summed/max(cnt,1)) @ Wl^T + b + h @ Wr^T   [+ relu]
// using V_WMMA_F32_16X16X4_F32 over K=64 (16 steps x 2 wmma each).
// Block = 128 threads = 4 waves -> one block covers all 64 output features
// for a 16-node tile.
__global__ void __launch_bounds__(128)
sage_gemm_kernel(const float* __restrict__ h,
                 const float* __restrict__ summed,
                 const float* __restrict__ cnt,
                 const float* __restrict__ Wl,   // [64][64] row-major: Wl[o][k]
                 const float* __restrict__ bias, // [64]
                 const float* __restrict__ Wr,   // [64][64] row-major
                 float* __restrict__ out,
                 int relu) {
    const int lane  = threadIdx.x & 31;
    const int otile = threadIdx.x >> 5;   // 0..3 -> output-feature tile
    const int ntile = blockIdx.x;         // node tile (N/16 = 6250 exact)
    const int half  = lane >> 4;          // lane group 0/1 -> K sub-pair
    const int l16   = lane & 15;

    const int m = ntile * 16 + l16;       // A-matrix row (node)
    const int o = otile * 16 + l16;       // B-matrix column (output feature)

    // mean scale for this node row (isolated nodes: summed==0 anyway)
    const float scale = 1.0f / fmaxf(cnt[m], 1.0f);

    const float* __restrict__ aggRow = summed + (long long)m * F;
    const float* __restrict__ hRow   = h      + (long long)m * F;
    const float* __restrict__ wlRow  = Wl     + (long long)o * F;
    const float* __restrict__ wrRow  = Wr     + (long long)o * F;

    v8f c = {};
#pragma unroll
    for (int k0 = 0; k0 < F; k0 += 4) {
        const int ka = k0 + 2 * half;  // A/B VGPR layout: V0->K=2h, V1->K=2h+1
        // agg @ Wl^T
        v2f a1 = *(const v2f*)(aggRow + ka);
        a1 = a1 * scale;
        v2f b1 = *(const v2f*)(wlRow + ka);
        c = __builtin_amdgcn_wmma_f32_16x16x4_f32(
                /*neg_a=*/false, a1, /*neg_b=*/false, b1,
                /*c_mod=*/(short)0, c, /*reuse_a=*/false, /*reuse_b=*/false);
        // h @ Wr^T
        v2f a2 = *(const v2f*)(hRow + ka);
        v2f b2 = *(const v2f*)(wrRow + ka);
        c = __builtin_amdgcn_wmma_f32_16x16x4_f32(
                false, a2, false, b2, (short)0, c, false, false);
    }

    const float bv = bias[o];
#pragma unroll
    for (int r = 0; r < 8; ++r) {
        // C/D layout: VGPR r -> M = r (lanes 0-15) / r+8 (lanes 16-31), N = l16
        const int row = ntile * 16 + r + 8 * half;
        float v = c[r] + bv;
        if (relu) v = fmaxf(v, 0.0f);
        out[(long long)row * F + o] = v;
    }
}

extern "C" void kernel_launch(void* const* d_in, const int* in_sizes, int n_in,
                              void* d_out, int out_size, void* d_ws, size_t ws_size,
                              hipStream_t stream) {
    (void)in_sizes; (void)n_in; (void)out_size; (void)ws_size;

    const float* x   = (const float*)d_in[0];
    const int*   ei  = (const int*)  d_in[1];   // [2][E]: src row then dst row
    const float* W1l = (const float*)d_in[2];
    const float* b1  = (const float*)d_in[3];
    const float* W1r = (const float*)d_in[4];
    const float* W2l = (const float*)d_in[5];
    const float* b2  = (const float*)d_in[6];
    const float* W2r = (const float*)d_in[7];
    float* out = (float*)d_out;

    float* ws     = (float*)d_ws;
    float* summed = ws;                               // N*F floats
    float* cnt    = summed + (size_t)N_NODES * F;     // N floats
    float* h1     = cnt + N_NODES;                    // N*F floats

    const int* src = ei;
    const int* dst = ei + N_EDGES;

    // zero accumulator + degree counter every launch (deterministic replays)
    hipMemsetAsync(summed, 0, ((size_t)N_NODES * F + N_NODES) * sizeof(float), stream);

    const long long totalScatter = (long long)N_EDGES * F;
    const int tpb = 256;
    const int sblocks = (int)((totalScatter + tpb - 1) / tpb);
    const int gblocks = N_NODES / 16;  // 6250, exact

    // Layer 1
    sage_scatter_kernel<<<sblocks, tpb, 0, stream>>>(x, src, dst, summed, cnt, 1);
    sage_gemm_kernel<<<gblocks, 128, 0, stream>>>(x, summed, cnt, W1l, b1, W1r, h1, 1);

    // Layer 2 (reuse cnt; re-zero the accumulator mid-graph)
    hipMemsetAsync(summed, 0, (size_t)N_NODES * F * sizeof(float), stream);
    sage_scatter_kernel<<<sblocks, tpb, 0, stream>>>(h1, src, dst, summed, cnt, 0);
    sage_gemm_kernel<<<gblocks, 128, 0, stream>>>(h1, summed, cnt, W2l, b2, W2r, out, 0);
}